// GraphSAGERecommender_64819646431531
// MI455X (gfx1250) — compile-verified
//
#include <hip/hip_runtime.h>

// GraphSAGE recommender, MI455X (gfx1250, wave32).
// Pipeline per call:
//   zero(deg), zero(msg1=d_out)
//   scatter1: msg1[dst] += x[src] (64 ch) + degree count
//   gemm1:    h1 = relu([msg1/deg | x] @ [W1l|W1r].T + b1)   (K=128, fp32 WMMA)
//   zero(msg2)
//   scatter2: msg2[dst] += h1[src] (128 ch)
//   gemm2:    out = [msg2/deg | h1] @ [W2l|W2r].T + b2       (K=256, fp32 WMMA)

#define N_USERS  100000
#define N_MOVIES 50000
#define N_NODES  150000
#define N_EDGES  1000000
#define IN_CH    64
#define HID      128
#define OUT_CH   64

typedef __attribute__((ext_vector_type(2))) float v2f;
typedef __attribute__((ext_vector_type(8))) float v8f;

__device__ __forceinline__ void atomAddF(float* p, float v) {
  // agent-scope relaxed fp32 add -> global_atomic_add_f32 (no CAS loop)
  __hip_atomic_fetch_add(p, v, __ATOMIC_RELAXED, __HIP_MEMORY_SCOPE_AGENT);
}

__global__ void fill_zero_kernel(float* __restrict__ p, size_t n) {
  size_t i = (size_t)blockIdx.x * blockDim.x + threadIdx.x;
  size_t stride = (size_t)gridDim.x * blockDim.x;
  for (; i < n; i += stride) p[i] = 0.0f;
}

// One lane per (edge, channel). Coalesced gather of source row, f32 atomics into msg.
// SPLIT: source features come from two tables (users then movies).
template<int CH, bool SPLIT, bool DO_DEG>
__global__ void scatter_kernel(const int* __restrict__ ei,
                               const float* __restrict__ xa,
                               const float* __restrict__ xb,
                               float* __restrict__ msg,
                               float* __restrict__ deg) {
  constexpr int EPB = 256 / CH;
  const int c  = threadIdx.x % CH;
  const int le = threadIdx.x / CH;
  const long e = (long)blockIdx.x * EPB + le;
  if (e >= N_EDGES) return;
  const int src = ei[e];
  const int dst = ei[N_EDGES + e];
  const float* row;
  if (SPLIT)
    row = (src < N_USERS) ? xa + (size_t)src * CH
                          : xb + (size_t)(src - N_USERS) * CH;
  else
    row = xa + (size_t)src * CH;
  atomAddF(msg + (size_t)dst * CH + c, row[c]);
  if (DO_DEG && c == 0) atomAddF(deg + dst, 1.0f);
}

// Fused SAGE layer GEMM using V_WMMA_F32_16X16X4_F32.
//   out[n, :] = act( [msg[n]/max(deg,1) | x[n]] @ Bc + bias ),  Bc[k][h] = [Wl|Wr]
// Block = (D_OUT/16) waves; each wave owns one 16-wide N tile; loop over MT
// M-tiles of 16 nodes. K is processed in two halves so the staged B slice is
// only 32 KB of LDS (plus the padded 16xK A tile).
template<int D_IN, int D_OUT, bool RELU, bool SPLIT, int MT>
__global__ void sage_gemm_kernel(const float* __restrict__ xa,
                                 const float* __restrict__ xb,
                                 const float* __restrict__ msg,
                                 const float* __restrict__ deg,
                                 const float* __restrict__ Wl,
                                 const float* __restrict__ Wr,
                                 const float* __restrict__ bias,
                                 float* __restrict__ out) {
  constexpr int K    = 2 * D_IN;   // concat [agg | x]
  constexpr int KH   = K / 2;      // staged K-half
  constexpr int ASTR = K + 4;      // +4 floats: conflict-free half-wave A reads
  constexpr int NW   = D_OUT / 16; // waves per block

  __shared__ float Bls[KH * D_OUT]; // 32 KB (both layers)
  __shared__ float Als[16 * ASTR];

  const int tid      = threadIdx.x;
  const int nthreads = NW * 32;
  const int lane  = tid & 31;
  const int wid   = tid >> 5;
  const int n0    = wid * 16;         // this wave's N tile
  const int mrow  = lane & 15;
  const int khalf = (lane >> 4) * 2;  // ISA 16x4 f32 layout: hi half-wave = K+2/K+3

  for (int t = 0; t < MT; ++t) {
    const long tile  = (long)blockIdx.x * MT + t;
    const long node0 = tile * 16;

    // Stage A tile: [agg | x] rows (coalesced along k)
    for (int idx = tid; idx < 16 * K; idx += nthreads) {
      const int r = idx / K;
      const int k = idx % K;
      const long node = node0 + r;
      float v;
      if (k < D_IN) {
        const float d = deg[node];
        v = msg[node * D_IN + k] * (1.0f / fmaxf(d, 1.0f));
      } else {
        const int kk = k - D_IN;
        if (SPLIT)
          v = (node < N_USERS) ? xa[node * D_IN + kk]
                               : xb[(node - N_USERS) * D_IN + kk];
        else
          v = xa[node * D_IN + kk];
      }
      Als[r * ASTR + k] = v;
    }
    __syncthreads();

    v8f acc = {};
    #pragma unroll
    for (int kb = 0; kb < K; kb += KH) {
      // Stage B half: Bls[k][h] = (kb+k < D_IN) ? Wl[h][kb+k] : Wr[h][kb+k-D_IN]
      for (int idx = tid; idx < KH * D_OUT; idx += nthreads) {
        const int k  = idx % KH;
        const int h  = idx / KH;
        const int gk = kb + k;
        const float w = (gk < D_IN) ? Wl[h * D_IN + gk]
                                    : Wr[h * D_IN + (gk - D_IN)];
        Bls[k * D_OUT + h] = w;
      }
      __syncthreads();

      #pragma unroll 8
      for (int k0 = 0; k0 < KH; k0 += 4) {
        v2f a, b;
        a.x = Als[mrow * ASTR + kb + k0 + khalf];
        a.y = Als[mrow * ASTR + kb + k0 + khalf + 1];
        b.x = Bls[(k0 + khalf) * D_OUT + n0 + mrow];
        b.y = Bls[(k0 + khalf + 1) * D_OUT + n0 + mrow];
        acc = __builtin_amdgcn_wmma_f32_16x16x4_f32(
            /*neg_a=*/false, a, /*neg_b=*/false, b,
            /*c_mod=*/(short)0, acc, /*reuse_a=*/false, /*reuse_b=*/false);
      }
      __syncthreads();
    }

    // D layout: VGPR i holds M=i (lanes 0-15) and M=i+8 (lanes 16-31), N=lane&15
    const int rbase = (lane >> 4) * 8;
    const int col   = n0 + mrow;
    const float bv  = bias[col];
    #pragma unroll
    for (int i = 0; i < 8; ++i) {
      float v = acc[i] + bv;
      if (RELU) v = fmaxf(v, 0.0f);
      out[(node0 + rbase + i) * D_OUT + col] = v;
    }
    __syncthreads();
  }
}

extern "C" void kernel_launch(void* const* d_in, const int* in_sizes, int n_in,
                              void* d_out, int out_size, void* d_ws, size_t ws_size,
                              hipStream_t stream) {
  const int*   ei   = (const int*)d_in[0];
  const float* uemb = (const float*)d_in[1];
  const float* memb = (const float*)d_in[2];
  const float* W1l  = (const float*)d_in[3];
  const float* b1   = (const float*)d_in[4];
  const float* W1r  = (const float*)d_in[5];
  const float* W2l  = (const float*)d_in[6];
  const float* b2   = (const float*)d_in[7];
  const float* W2r  = (const float*)d_in[8];
  float* out = (float*)d_out;

  // Workspace plan (d_ws): deg [N] | h1 [N*128] | msg2 [N*128]  (~155 MB)
  float* ws   = (float*)d_ws;
  float* deg  = ws;
  float* h1   = ws + (1 << 18);                 // 1 MiB-aligned region
  float* msg2 = h1 + (size_t)N_NODES * HID;
  float* msg1 = out;                            // reuse d_out as layer-1 accumulator

  fill_zero_kernel<<<1024, 256, 0, stream>>>(deg, (size_t)N_NODES);
  fill_zero_kernel<<<4096, 256, 0, stream>>>(msg1, (size_t)N_NODES * IN_CH);

  // Layer 1 scatter (+degree): 4 edges/block, 64 lanes per edge
  scatter_kernel<IN_CH, true, true>
      <<<N_EDGES / (256 / IN_CH), 256, 0, stream>>>(ei, uemb, memb, msg1, deg);

  // Layer 1 fused GEMM+bias+relu: 9375 M-tiles = 1875 blocks x 5 tiles, 8 waves
  sage_gemm_kernel<IN_CH, HID, true, true, 5>
      <<<1875, 256, 0, stream>>>(uemb, memb, msg1, deg, W1l, W1r, b1, h1);

  fill_zero_kernel<<<8192, 256, 0, stream>>>(msg2, (size_t)N_NODES * HID);

  // Layer 2 scatter: 2 edges/block, 128 lanes per edge
  scatter_kernel<HID, false, false>
      <<<N_EDGES / (256 / HID), 256, 0, stream>>>(ei, h1, nullptr, msg2, nullptr);

  // Layer 2 fused GEMM+bias: 4 waves per block
  sage_gemm_kernel<HID, OUT_CH, false, false, 5>
      <<<1875, 128, 0, stream>>>(h1, nullptr, msg2, deg, W2l, W2r, b2, out);
}